// DPP_83519934038264
// MI455X (gfx1250) — compile-verified
//
#include <hip/hip_runtime.h>
#include <hip/hip_bf16.h>
#include <math.h>

typedef __attribute__((ext_vector_type(2))) float v2f;
typedef __attribute__((ext_vector_type(8))) float v8f;

#define NDIM   2048   // n
#define KDIM   2000   // k
#define NBATCH 8
#define NMAT   (NBATCH + 1)   // 8 masked matrices + 1 normalization matrix
#define NB     64             // Cholesky panel width

#define WMMA_F32(A, B, C) \
  __builtin_amdgcn_wmma_f32_16x16x4_f32(false, (A), false, (B), (short)0, (C), false, false)

// ---------------------------------------------------------------------------
// G = B^T B with V_WMMA_F32_16X16X4_F32, 2x2 register-blocked: one wave owns a
// 32x32 output tile (4 accumulators, 32 VGPRs), so each A/B fragment feeds two
// WMMAs -> half the operand traffic per FLOP vs 16x16/wave. All loads are
// coalesced (16 consecutive floats per half-wave). B (16 MiB) is L2-resident
// on MI455X (192 MB L2), so the re-streaming hits L2, not HBM.
//
// Fragment layouts (ISA 7.12.2, 32-bit): A 16x4: lanes 0-15 hold K=0,1 in the
// two VGPRs, lanes 16-31 hold K=2,3; M = lane&15. B 4x16 mirrored, N = lane&15.
// C/D 16x16: VGPR v -> row v (+8 for lanes 16-31), col = lane&15.
// ---------------------------------------------------------------------------
__global__ __launch_bounds__(32) void gram_kernel(const float* __restrict__ B,
                                                  float* __restrict__ G) {
  const int lane = threadIdx.x;
  const int half = lane >> 4;
  const int lr   = lane & 15;
  const int i0   = blockIdx.y * 32;
  const int j0   = blockIdx.x * 32;
  const int kkb  = half * 2;

  v8f acc00 = {}, acc01 = {}, acc10 = {}, acc11 = {};
  for (int k0 = 0; k0 < KDIM; k0 += 4) {
    const float* r0p = B + (size_t)(k0 + kkb) * NDIM;
    const float* r1p = r0p + NDIM;
    v2f a0, a1, b0, b1;
    a0.x = r0p[i0 + lr];       a0.y = r1p[i0 + lr];
    a1.x = r0p[i0 + 16 + lr];  a1.y = r1p[i0 + 16 + lr];
    b0.x = r0p[j0 + lr];       b0.y = r1p[j0 + lr];
    b1.x = r0p[j0 + 16 + lr];  b1.y = r1p[j0 + 16 + lr];
    acc00 = WMMA_F32(a0, b0, acc00);
    acc01 = WMMA_F32(a0, b1, acc01);
    acc10 = WMMA_F32(a1, b0, acc10);
    acc11 = WMMA_F32(a1, b1, acc11);
  }

  const int c0 = j0 + lr, c1 = j0 + 16 + lr;
  const int ra = i0 + half * 8, rb = i0 + 16 + half * 8;
#pragma unroll
  for (int v = 0; v < 8; ++v) {
    G[(size_t)(ra + v) * NDIM + c0] = acc00[v];
    G[(size_t)(ra + v) * NDIM + c1] = acc01[v];
    G[(size_t)(rb + v) * NDIM + c0] = acc10[v];
    G[(size_t)(rb + v) * NDIM + c1] = acc11[v];
  }
}

// ---------------------------------------------------------------------------
// Build matrix (mat_base + blockIdx.z):
//   mat < NBATCH : M = mask_i*mask_j*G + diag(1-mask)
//   mat == NBATCH: M = G + I   (Sylvester: det(I_K + B B^T) = det(I_N + G))
// ---------------------------------------------------------------------------
__global__ void build_kernel(const float* __restrict__ G, float* __restrict__ Mbase,
                             size_t mstride, const int* __restrict__ x, int mat_base) {
  size_t gid = (size_t)blockIdx.x * blockDim.x + threadIdx.x;
  if (gid >= (size_t)NDIM * NDIM) return;
  const int mat = mat_base + blockIdx.z;
  float* M = Mbase + (size_t)blockIdx.z * mstride;
  const int* xr = (mat < NBATCH) ? (x + (size_t)mat * NDIM) : nullptr;

  int i = (int)(gid / NDIM), j = (int)(gid % NDIM);
  float mi = 1.0f, mj = 1.0f, dadd = 1.0f;
  if (xr) {
    mi = (xr[i] == 1) ? 1.0f : 0.0f;
    mj = (xr[j] == 1) ? 1.0f : 0.0f;
    dadd = 1.0f - mi;
  }
  float v = G[gid] * (mi * mj);
  if (i == j) v += dadd;
  M[gid] = v;
}

// ---------------------------------------------------------------------------
// Factor the NB x NB diagonal block in LDS (64 threads per matrix, one block
// per blockIdx.z), write L back, accumulate 2*sum(log(diag)).
// ---------------------------------------------------------------------------
__global__ __launch_bounds__(64) void potf_kernel(float* __restrict__ Mbase,
                                                  size_t mstride, int k0,
                                                  float* __restrict__ accbase) {
  __shared__ float As[NB][NB + 1];
  float* M = Mbase + (size_t)blockIdx.z * mstride;
  const int tid = threadIdx.x;

  for (int idx = tid; idx < NB * NB; idx += 64) {
    int r = idx >> 6, c = idx & 63;
    As[r][c] = M[(size_t)(k0 + r) * NDIM + (k0 + c)];
  }
  __syncthreads();
  for (int j = 0; j < NB; ++j) {
    if (tid == j) As[j][j] = sqrtf(As[j][j]);
    __syncthreads();
    float inv = 1.0f / As[j][j];
    if (tid > j) As[tid][j] *= inv;
    __syncthreads();
    if (tid > j) {
      float lij = As[tid][j];
      for (int c = j + 1; c <= tid; ++c) As[tid][c] -= lij * As[c][j];
    }
    __syncthreads();
  }
  for (int idx = tid; idx < NB * NB; idx += 64) {
    int r = idx >> 6, c = idx & 63;
    M[(size_t)(k0 + r) * NDIM + (k0 + c)] = As[r][c];
  }
  if (tid == 0) {
    float s = 0.0f;
    for (int j = 0; j < NB; ++j) s += 2.0f * logf(As[j][j]);
    atomicAdd(accbase + blockIdx.z, s);
  }
}

// ---------------------------------------------------------------------------
// Panel solve: rows i >= k0+NB, X * Lkk^T = A (forward substitution over NB
// columns). One thread per row; Lkk + per-thread row in LDS.
// ---------------------------------------------------------------------------
__global__ __launch_bounds__(64) void trsm_kernel(float* __restrict__ Mbase,
                                                  size_t mstride, int k0) {
  __shared__ float Lkk[NB][NB + 1];
  __shared__ float Xs[64][NB + 1];
  float* M = Mbase + (size_t)blockIdx.z * mstride;
  const int tid = threadIdx.x;
  const int row = k0 + NB + blockIdx.x * 64 + tid;   // grid sized so row < NDIM

  for (int idx = tid; idx < NB * NB; idx += 64) {
    int r = idx >> 6, c = idx & 63;
    Lkk[r][c] = M[(size_t)(k0 + r) * NDIM + (k0 + c)];
  }
  __syncthreads();

  float* xr = &Xs[tid][0];
  for (int c = 0; c < NB; ++c) xr[c] = M[(size_t)row * NDIM + (k0 + c)];
  for (int c = 0; c < NB; ++c) {
    float s = xr[c];
    for (int t = 0; t < c; ++t) s -= xr[t] * Lkk[c][t];
    xr[c] = s / Lkk[c][c];
  }
  for (int c = 0; c < NB; ++c) M[(size_t)row * NDIM + (k0 + c)] = xr[c];
}

// ---------------------------------------------------------------------------
// Trailing update (SYRK): M[r,c] -= sum_k L[r,k]*L[c,k] over the NB panel.
// 2x2 register-blocked: one wave owns a 32x32 tile (64 WMMAs per tile,
// operands shared across pairs). Lower-triangle 32-blocks only; spurious
// strictly-upper writes inside diagonal 32-blocks are never read again.
// ---------------------------------------------------------------------------
__global__ __launch_bounds__(32) void syrk_kernel(float* __restrict__ Mbase,
                                                  size_t mstride, int k0) {
  const int k1 = k0 + NB;
  const int r0 = k1 + blockIdx.y * 32;
  const int c0 = k1 + blockIdx.x * 32;
  if (c0 > r0) return;                 // block-uniform: whole wave exits
  float* M = Mbase + (size_t)blockIdx.z * mstride;

  const int lane = threadIdx.x;
  const int half = lane >> 4;
  const int lr   = lane & 15;
  const int kkb  = half * 2;

  const float* Pa0 = M + (size_t)(r0 + lr) * NDIM + k0;
  const float* Pa1 = Pa0 + (size_t)16 * NDIM;
  const float* Pb0 = M + (size_t)(c0 + lr) * NDIM + k0;
  const float* Pb1 = Pb0 + (size_t)16 * NDIM;

  v8f acc00 = {}, acc01 = {}, acc10 = {}, acc11 = {};
  for (int kk = 0; kk < NB; kk += 4) {
    v2f a0, a1, b0, b1;
    a0.x = Pa0[kk + kkb];  a0.y = Pa0[kk + kkb + 1];
    a1.x = Pa1[kk + kkb];  a1.y = Pa1[kk + kkb + 1];
    b0.x = Pb0[kk + kkb];  b0.y = Pb0[kk + kkb + 1];
    b1.x = Pb1[kk + kkb];  b1.y = Pb1[kk + kkb + 1];
    acc00 = WMMA_F32(a0, b0, acc00);
    acc01 = WMMA_F32(a0, b1, acc01);
    acc10 = WMMA_F32(a1, b0, acc10);
    acc11 = WMMA_F32(a1, b1, acc11);
  }

  const int col0 = c0 + lr, col1 = c0 + 16 + lr;
  const int ra = r0 + half * 8, rb = r0 + 16 + half * 8;
#pragma unroll
  for (int v = 0; v < 8; ++v) {
    M[(size_t)(ra + v) * NDIM + col0] -= acc00[v];
    M[(size_t)(ra + v) * NDIM + col1] -= acc01[v];
    M[(size_t)(rb + v) * NDIM + col0] -= acc10[v];
    M[(size_t)(rb + v) * NDIM + col1] -= acc11[v];
  }
}

__global__ void init_kernel(float* __restrict__ acc) {
  if (threadIdx.x < NMAT) acc[threadIdx.x] = 0.0f;
}

__global__ void finish_kernel(const float* __restrict__ acc,
                              float* __restrict__ out) {
  int b = threadIdx.x;
  if (b < NBATCH) out[b] = acc[b] - acc[NBATCH];
}

// ---------------------------------------------------------------------------
// Orchestration. Preferred (batched) workspace layout:
//   G (16 MiB) | M[0..8] (9 x 16 MiB) | acc (9 floats)   ~= 160 MiB
// All 9 independent Choleskys advance in lockstep via blockIdx.z, cutting the
// serialized launch chain ~9x (the latency-bound part of this workload).
// If ws_size is too small, fall back to one shared M buffer processed
// sequentially. Both paths are deterministic for a fixed ws_size.
// ---------------------------------------------------------------------------
extern "C" void kernel_launch(void* const* d_in, const int* in_sizes, int n_in,
                              void* d_out, int out_size, void* d_ws, size_t ws_size,
                              hipStream_t stream) {
  const int*   x = (const int*)d_in[0];     // [NBATCH, NDIM] int32 mask
  const float* B = (const float*)d_in[1];   // [KDIM, NDIM] float32
  float*       out = (float*)d_out;         // [NBATCH] float32

  const size_t NN = (size_t)NDIM * NDIM;
  const int    build_gx = (int)((NN + 255) / 256);
  float* G = (float*)d_ws;

  const size_t need_batched = (NN * (size_t)(1 + NMAT) + 64) * sizeof(float);

  if (ws_size >= need_batched) {
    float* Mb  = G + NN;             // 9 matrices, stride NN
    float* acc = Mb + (size_t)NMAT * NN;

    init_kernel<<<1, 32, 0, stream>>>(acc);
    gram_kernel<<<dim3(NDIM / 32, NDIM / 32), 32, 0, stream>>>(B, G);
    build_kernel<<<dim3(build_gx, 1, NMAT), 256, 0, stream>>>(G, Mb, NN, x, 0);

    for (int k0 = 0; k0 < NDIM; k0 += NB) {
      potf_kernel<<<dim3(1, 1, NMAT), 64, 0, stream>>>(Mb, NN, k0, acc);
      int below = NDIM - k0 - NB;
      if (below > 0) {
        trsm_kernel<<<dim3(below / 64, 1, NMAT), 64, 0, stream>>>(Mb, NN, k0);
        syrk_kernel<<<dim3(below / 32, below / 32, NMAT), 32, 0, stream>>>(Mb, NN, k0);
      }
    }
    finish_kernel<<<1, 32, 0, stream>>>(acc, out);
  } else {
    // Fallback: G | M | acc  (~32 MiB), matrices processed sequentially.
    float* M   = G + NN;
    float* acc = M + NN;

    init_kernel<<<1, 32, 0, stream>>>(acc);
    gram_kernel<<<dim3(NDIM / 32, NDIM / 32), 32, 0, stream>>>(B, G);

    for (int mat = 0; mat < NMAT; ++mat) {
      build_kernel<<<dim3(build_gx, 1, 1), 256, 0, stream>>>(G, M, 0, x, mat);
      for (int k0 = 0; k0 < NDIM; k0 += NB) {
        potf_kernel<<<dim3(1, 1, 1), 64, 0, stream>>>(M, 0, k0, acc + mat);
        int below = NDIM - k0 - NB;
        if (below > 0) {
          trsm_kernel<<<dim3(below / 64, 1, 1), 64, 0, stream>>>(M, 0, k0);
          syrk_kernel<<<dim3(below / 32, below / 32, 1), 32, 0, stream>>>(M, 0, k0);
        }
      }
    }
    finish_kernel<<<1, 32, 0, stream>>>(acc, out);
  }
}